// DSVF_47725676593729
// MI455X (gfx1250) — compile-verified
//
#include <hip/hip_runtime.h>
#include <math.h>

// ---------------------------------------------------------------------------
// Biquad IIR over [B=32, T=2^20] via 3-pass block-parallel scan + WMMA.
//
// DF-I view:  y_t = b0 x_t + b1 x_{t-1} + b2 x_{t-2} - a1 y_{t-1} - a2 y_{t-2}
// History enters linearly through forcing terms
//   F0 = b1 x_{-1} + b2 x_{-2} - a1 y_{-1} - a2 y_{-2}
//   F1 = b2 x_{-1} - a2 y_{-1}
//   y[t] = y_zero_history[t] + psi[t] F0 + psi[t-1] F1
// where psi solves  psi[t] = delta[t] - a1 psi[t-1] - a2 psi[t-2].
// Zero-history output of a 16-sample block is H x (H = lower-tri Toeplitz of
// the impulse response h = conv(psi, [b0,b1,b2])).  16 blocks at a time are
// computed as a 16x16x16 matmul via 4x V_WMMA_F32_16X16X4_F32.
// ---------------------------------------------------------------------------

typedef float v2f __attribute__((ext_vector_type(2)));
typedef float v4f __attribute__((ext_vector_type(4)));
typedef float v8f __attribute__((ext_vector_type(8)));

#define T_LEN   1048576
#define BATCH   32
#define S_CHUNK 4096              // samples per superchunk (one wave)
#define NSC     (T_LEN / S_CHUNK) // 256 superchunks per row
#define TILES   16                // 256-sample WMMA tiles per superchunk
#define D_OFF   64                                  // ws float offsets
#define HIST_OFF (64 + BATCH * NSC * 2)
#define WAVE_LDS 384              // 320 transpose words + 64 scan words
#define PI_F 3.14159265358979323846f

// ---------------------------------------------------------------------------
// Setup: compute filter coefficients + psi/h tables on device.
// ---------------------------------------------------------------------------
__global__ void k_setup(const float* g, const float* r, const float* m_hp,
                        const float* m_bp, const float* m_lp, float* ws) {
  if (threadIdx.x != 0 || blockIdx.x != 0) return;
  float sig = 1.0f / (1.0f + expf(-g[0]));
  float gg  = tanf(PI_F * sig * 0.5f);
  float rr  = (r[0] > 20.0f) ? r[0] : log1pf(expf(r[0]));   // softplus
  float g2  = gg * gg;
  float hp = m_hp[0], bp = m_bp[0], lp = m_lp[0];
  float b0 = g2 * lp + gg * bp + hp;
  float b1 = 2.0f * g2 * lp - 2.0f * hp;
  float b2 = g2 * lp - gg * bp + hp;
  float a0 = g2 + 2.0f * rr * gg + 1.0f;
  float a1 = 2.0f * g2 - 2.0f;
  float a2 = g2 - 2.0f * rr * gg + 1.0f;
  float inv = 1.0f / a0;
  b0 *= inv; b1 *= inv; b2 *= inv; a1 *= inv; a2 *= inv;
  ws[0] = b0; ws[1] = b1; ws[2] = b2; ws[3] = a1; ws[4] = a2;

  float psi[16];
  psi[0] = 1.0f;
  psi[1] = -a1;
  for (int t = 2; t < 16; ++t) psi[t] = -a1 * psi[t - 1] - a2 * psi[t - 2];
  for (int t = 0; t < 16; ++t) {
    ws[8 + t] = psi[t];
    float h = b0 * psi[t];
    if (t >= 1) h += b1 * psi[t - 1];
    if (t >= 2) h += b2 * psi[t - 2];
    ws[24 + t] = h;                       // impulse response h[0..15]
  }
  // continue psi to t = S-1 for the superchunk-boundary scan
  float pa = psi[15], pb = psi[14], pc = psi[13];
  for (int t = 16; t < S_CHUNK; ++t) {
    float pn = -a1 * pa - a2 * pb;
    pc = pb; pb = pa; pa = pn;
  }
  ws[5] = pa;  // psi[S-1]
  ws[6] = pb;  // psi[S-2]
  ws[7] = pc;  // psi[S-3]
}

// ---------------------------------------------------------------------------
// Per-wave superchunk worker (shared by pass 1 and pass 3).
// WMMA fragment layouts per CDNA5 ISA 7.12.2 (wave32, f32):
//   A 16x4 : lane = c*16+m holds {H[m][4q+2c], H[m][4q+2c+1]} for K-slice q
//   B 4x16 : lane = c*16+n holds {X[4q+2c][n], X[4q+2c+1][n]}
//   C 16x16: vgpr r, lane = c*16+n  ->  element (m = 8c+r, n)
// X[k][n] = sample n*16+k of the tile (column n = block n).
// ---------------------------------------------------------------------------
template <bool WRITE>
__device__ __forceinline__ void run_superchunk(const float* __restrict__ x,
                                               float* __restrict__ y,
                                               float* ws, float* wls,
                                               int row, int sc, int lane) {
  const float* hdr = ws;
  const int c  = lane >> 4;      // half-wave select
  const int nn = lane & 15;      // row (A) / column (B,C) index

  const float b0 = hdr[0], b1 = hdr[1], b2 = hdr[2];
  const float a1 = hdr[3], a2 = hdr[4];
  const float psi13 = hdr[8 + 13], psi14 = hdr[8 + 14], psi15 = hdr[8 + 15];
  (void)b0;

  // per-lane psi selections for the rank-2 C-tile update: m = 8c + r
  float pm[8], pm1[8];
#pragma unroll
  for (int r = 0; r < 8; ++r) {
    pm[r]  = hdr[8 + 8 * c + r];
    pm1[r] = (r == 0) ? (c ? hdr[8 + 7] : 0.0f) : hdr[8 + 8 * c + r - 1];
  }

  // A fragments: H[m][k] = h[m-k] (lower-triangular Toeplitz)
  v2f afr[4];
#pragma unroll
  for (int q = 0; q < 4; ++q) {
    int i0 = nn - (4 * q + 2 * c);
    afr[q].x = (i0 >= 0) ? hdr[24 + i0] : 0.0f;
    afr[q].y = (i0 >= 1) ? hdr[24 + i0 - 1] : 0.0f;
  }

  const float* xsc = x + (size_t)row * T_LEN + (size_t)sc * S_CHUNK;
  float* ysc = WRITE ? (y + (size_t)row * T_LEN + (size_t)sc * S_CHUNK) : nullptr;

  // carried history (x_{-1}, x_{-2}, y_{-1}, y_{-2})
  float xm1 = 0.0f, xm2 = 0.0f, ym1 = 0.0f, ym2 = 0.0f;
  if (WRITE) {
    const float* hist = ws + HIST_OFF + (size_t)(row * NSC + sc) * 2;
    ym1 = hist[0];
    ym2 = hist[1];
    if (sc > 0) {
      xm1 = xsc[-1];
      xm2 = xsc[-2];
    }
  }

  for (int t = 0; t < TILES; ++t) {
    const float* tb = xsc + t * 256;
    if (t < TILES - 1)
      __builtin_prefetch(tb + 256 + lane * 8, 0, 0);  // global_prefetch_b8

    // B fragments: two consecutive samples per lane per K-slice
    v2f bfr[4];
#pragma unroll
    for (int q = 0; q < 4; ++q)
      bfr[q] = *(const v2f*)(tb + nn * 16 + 4 * q + 2 * c);

    // Yz = H * X  (16x16x16 = 4 chained 16x16x4 f32 WMMAs)
    v8f acc = (v8f)0.0f;
#pragma unroll
    for (int q = 0; q < 4; ++q)
      acc = __builtin_amdgcn_wmma_f32_16x16x4_f32(
          false, afr[q], false, bfr[q], (short)0, acc, false, false);

    // publish per-block scan tuple {x14, x15, Yz14, Yz15}; K=14,15 live in
    // bfr[3] of the c==1 half-wave, rows 14/15 of C live in vgprs 6/7 there.
    if (c) {
      v4f tpl = {bfr[3].x, bfr[3].y, acc[6], acc[7]};
      *(v4f*)&wls[320 + 4 * nn] = tpl;
    }
    __builtin_amdgcn_wave_barrier();   // DS ops are in-order within a wave

    // redundant wave-uniform 16-block history scan; each lane latches the
    // forcing terms of its own column.
    float F0n = 0.0f, F1n = 0.0f;
#pragma unroll
    for (int j = 0; j < 16; ++j) {
      v4f tpl = *(const v4f*)&wls[320 + 4 * j];
      float F0 = b1 * xm1 + b2 * xm2 - a1 * ym1 - a2 * ym2;
      float F1 = b2 * xm1 - a2 * ym1;
      if (j == nn) { F0n = F0; F1n = F1; }
      float y14 = tpl.z + psi14 * F0 + psi13 * F1;
      float y15 = tpl.w + psi15 * F0 + psi14 * F1;
      xm2 = tpl.x; xm1 = tpl.y;
      ym2 = y14;   ym1 = y15;
    }

    if (WRITE) {
      // rank-2 history correction of the whole C tile
      float yf[8];
#pragma unroll
      for (int r = 0; r < 8; ++r)
        yf[r] = acc[r] + pm[r] * F0n + pm1[r] * F1n;

      // transpose (m,n)->time-major through LDS, then coalesced b128 stores
      v4f lo = {yf[0], yf[1], yf[2], yf[3]};
      v4f hi = {yf[4], yf[5], yf[6], yf[7]};
      *(v4f*)&wls[20 * nn + 8 * c]     = lo;   // block nn, samples 8c..8c+3
      *(v4f*)&wls[20 * nn + 8 * c + 4] = hi;   // block nn, samples 8c+4..8c+7
      __builtin_amdgcn_wave_barrier();
      int j2 = lane >> 1, hf = lane & 1;
      v4f o0 = *(const v4f*)&wls[20 * j2 + 8 * hf];
      v4f o1 = *(const v4f*)&wls[20 * j2 + 8 * hf + 4];
      *(v4f*)(ysc + t * 256 + 8 * lane)     = o0;
      *(v4f*)(ysc + t * 256 + 8 * lane + 4) = o1;
      __builtin_amdgcn_wave_barrier();
    }
  }

  if (!WRITE && lane == 0) {
    float* d = ws + D_OFF + (size_t)(row * NSC + sc) * 2;
    d[0] = ym2;  // zero-history y[S-2]
    d[1] = ym1;  // zero-history y[S-1]
  }
}

__global__ void __launch_bounds__(256) k_pass1(const float* __restrict__ x,
                                               float* ws) {
  __shared__ float lds[8 * WAVE_LDS];
  int wave = blockIdx.x * 8 + (threadIdx.x >> 5);
  int lane = threadIdx.x & 31;
  float* wls = &lds[(threadIdx.x >> 5) * WAVE_LDS];
  run_superchunk<false>(x, nullptr, ws, wls, wave >> 8, wave & (NSC - 1), lane);
}

__global__ void __launch_bounds__(256) k_pass3(const float* __restrict__ x,
                                               float* __restrict__ y,
                                               float* ws) {
  __shared__ float lds[8 * WAVE_LDS];
  int wave = blockIdx.x * 8 + (threadIdx.x >> 5);
  int lane = threadIdx.x & 31;
  float* wls = &lds[(threadIdx.x >> 5) * WAVE_LDS];
  run_superchunk<true>(x, y, ws, wls, wave >> 8, wave & (NSC - 1), lane);
}

// ---------------------------------------------------------------------------
// Pass 2: sequential scan over superchunk boundaries (one thread per row).
// hist[k] = y-history (y_{-1}, y_{-2}) at the START of superchunk k.
// ---------------------------------------------------------------------------
__global__ void k_scan(const float* __restrict__ x, float* ws) {
  int row = threadIdx.x;  // launched with exactly 32 threads
  const float* hdr = ws;
  float b1 = hdr[1], b2 = hdr[2], a1 = hdr[3], a2 = hdr[4];
  float pS1 = hdr[5], pS2 = hdr[6], pS3 = hdr[7];
  const float* d = ws + D_OFF + (size_t)row * NSC * 2;
  float* hist = ws + HIST_OFF + (size_t)row * NSC * 2;
  const float* xrow = x + (size_t)row * T_LEN;

  float xm1 = 0.0f, xm2 = 0.0f, ym1 = 0.0f, ym2 = 0.0f;
  hist[0] = 0.0f;
  hist[1] = 0.0f;
  for (int k = 0; k < NSC - 1; ++k) {
    float F0 = b1 * xm1 + b2 * xm2 - a1 * ym1 - a2 * ym2;
    float F1 = b2 * xm1 - a2 * ym1;
    float yS1 = d[2 * k + 1] + pS1 * F0 + pS2 * F1;  // y[S-1] of chunk k
    float yS2 = d[2 * k + 0] + pS2 * F0 + pS3 * F1;  // y[S-2] of chunk k
    ym1 = yS1; ym2 = yS2;
    xm1 = xrow[(size_t)(k + 1) * S_CHUNK - 1];
    xm2 = xrow[(size_t)(k + 1) * S_CHUNK - 2];
    hist[2 * (k + 1) + 0] = ym1;
    hist[2 * (k + 1) + 1] = ym2;
  }
}

// ---------------------------------------------------------------------------
extern "C" void kernel_launch(void* const* d_in, const int* in_sizes, int n_in,
                              void* d_out, int out_size, void* d_ws,
                              size_t ws_size, hipStream_t stream) {
  const float* x    = (const float*)d_in[0];
  const float* g    = (const float*)d_in[1];
  const float* r    = (const float*)d_in[2];
  const float* m_hp = (const float*)d_in[3];
  const float* m_bp = (const float*)d_in[4];
  const float* m_lp = (const float*)d_in[5];
  float* out = (float*)d_out;
  float* ws  = (float*)d_ws;

  k_setup<<<1, 32, 0, stream>>>(g, r, m_hp, m_bp, m_lp, ws);
  k_pass1<<<(BATCH * NSC) / 8, 256, 0, stream>>>(x, ws);
  k_scan<<<1, 32, 0, stream>>>(x, ws);
  k_pass3<<<(BATCH * NSC) / 8, 256, 0, stream>>>(x, out, ws);
}